// SMPL_Layer_16003048145114
// MI455X (gfx1250) — compile-verified
//
#include <hip/hip_runtime.h>
#include <math.h>

#define BS   256
#define NVRT 10475
#define NJ   55
#define N3   (NVRT*3)      // 31425
#define KB   60            // active betas (10 shape + 50 expression)
#define KPF  488           // pose_feature padded from 486 to multiple of 4
#define PFS  492           // LDS row stride for pf tile (bank-conflict free)

__device__ __constant__ int c_par[NJ] = {
  -1,0,0,0,1,2,3,4,5,6,7,8,9,9,9,12,13,14,16,17,18,19,15,15,15,20,25,26,20,
  28,29,20,31,32,20,34,35,20,37,38,21,40,41,21,43,44,21,46,47,21,49,50,21,52,53};

typedef float v2f __attribute__((ext_vector_type(2)));
typedef float v8f __attribute__((ext_vector_type(8)));

static __device__ __forceinline__ int imin(int a, int b) { return a < b ? a : b; }

#define WMMA4(A_, B_, C_) \
  __builtin_amdgcn_wmma_f32_16x16x4_f32(false, (A_), false, (B_), (short)0, (C_), false, false)

// ---------------------------------------------------------------------------
// K1: Rodrigues -> rot mats (BS,55,9) + pose_feature (BS,488) zero-padded
// ---------------------------------------------------------------------------
__global__ void k_rodrigues(const float* __restrict__ pose,
                            const float* __restrict__ go,
                            float* __restrict__ rotm,
                            float* __restrict__ pf)
{
  int id = blockIdx.x * 256 + threadIdx.x;
  if (id >= BS * NJ) return;
  int b = id / NJ, i = id - b * NJ;
  float rx, ry, rz;
  if (i == 0)       { rx = go[0]; ry = go[1]; rz = go[2]; }
  else if (i <= 21) { const float* p = pose + (b*53 + i)*3;     rx = p[0]; ry = p[1]; rz = p[2]; }
  else if (i == 22) { const float* p = pose + (b*53 + 52)*3;    rx = p[0]; ry = p[1]; rz = p[2]; }
  else if (i <= 24) { rx = 0.f; ry = 0.f; rz = 0.f; }             // eyes
  else              { const float* p = pose + (b*53 + (i-3))*3; rx = p[0]; ry = p[1]; rz = p[2]; }

  float ex = rx + 1e-8f, ey = ry + 1e-8f, ez = rz + 1e-8f;
  float ang = sqrtf(ex*ex + ey*ey + ez*ez);
  float inv = 1.f / ang;
  float x = rx*inv, y = ry*inv, z = rz*inv;
  float s = sinf(ang), c = cosf(ang), t = 1.f - c;

  float R[9];
  R[0] = 1.f - t*(y*y + z*z); R[1] = -s*z + t*x*y;       R[2] =  s*y + t*x*z;
  R[3] =  s*z + t*x*y;        R[4] = 1.f - t*(x*x + z*z); R[5] = -s*x + t*y*z;
  R[6] = -s*y + t*x*z;        R[7] =  s*x + t*y*z;       R[8] = 1.f - t*(x*x + y*y);

  float* rm = rotm + (size_t)id * 9;
#pragma unroll
  for (int e = 0; e < 9; e++) rm[e] = R[e];

  if (i >= 1) {
    float* q = pf + (size_t)b * KPF + (i - 1) * 9;
#pragma unroll
    for (int e = 0; e < 9; e++) q[e] = R[e] - ((e == 0 || e == 4 || e == 8) ? 1.f : 0.f);
  } else {
    pf[(size_t)b * KPF + 486] = 0.f;
    pf[(size_t)b * KPF + 487] = 0.f;
  }
}

// ---------------------------------------------------------------------------
// K1b: pack active betas (shape 10 + expression 50) -> (BS,60)
// ---------------------------------------------------------------------------
__global__ void k_betas(const float* __restrict__ shp, const float* __restrict__ expr,
                        float* __restrict__ betas)
{
  int id = blockIdx.x * 256 + threadIdx.x;
  if (id >= BS * KB) return;
  int b = id / KB, l = id - b * KB;
  betas[id] = (l < 10) ? shp[b * 10 + l] : expr[b * 50 + (l - 10)];
}

// ---------------------------------------------------------------------------
// KT: transpose active shapedirs columns -> sdT[60][31425] (unit-stride B rows)
// ---------------------------------------------------------------------------
__global__ void k_sdT(const float* __restrict__ sd, float* __restrict__ sdT)
{
  int id = blockIdx.x * 256 + threadIdx.x;
  if (id >= KB * N3) return;
  int la = id / N3, n = id - la * N3;
  int col = (la < 10) ? la : (la + 290);
  sdT[id] = sd[(size_t)n * 350 + col];
}

// ---------------------------------------------------------------------------
// K2: JS[jc][la] = sum_v Jreg[j][v]*sdT[la][3v+c]; la==60 slot = Jreg@v_template
// ---------------------------------------------------------------------------
__global__ __launch_bounds__(128) void k_JS(const float* __restrict__ jreg,
                                            const float* __restrict__ sdT,
                                            const float* __restrict__ vt,
                                            float* __restrict__ JS)
{
  int jc = blockIdx.x;            // 0..164
  int j = jc / 3, cc = jc - j * 3;
  int t = threadIdx.x;            // 0..127
  float acc[61];
#pragma unroll
  for (int i = 0; i < 61; i++) acc[i] = 0.f;
  for (int v = t; v < NVRT; v += 128) {
    float wv = jreg[(size_t)j * NVRT + v];
    int n = v * 3 + cc;
#pragma unroll
    for (int la = 0; la < KB; la++)
      acc[la] += wv * sdT[(size_t)la * N3 + n];
    acc[60] += wv * vt[n];
  }
  __shared__ float red[61 * 128];
#pragma unroll
  for (int i = 0; i < 61; i++) red[i * 128 + t] = acc[i];
  __syncthreads();
  if (t < 61) {
    float s = 0.f;
    for (int i = 0; i < 128; i++) s += red[t * 128 + i];
    JS[jc * 61 + t] = s;
  }
}

// ---------------------------------------------------------------------------
// K3: rest joints[b][jc] = JS[jc][60] + sum_la betas[b][la]*JS[jc][la]
// ---------------------------------------------------------------------------
__global__ void k_joints(const float* __restrict__ betas, const float* __restrict__ JS,
                         float* __restrict__ joints)
{
  int id = blockIdx.x * 256 + threadIdx.x;
  if (id >= BS * 165) return;
  int b = id / 165, jc = id - b * 165;
  const float* js = JS + jc * 61;
  const float* be = betas + b * KB;
  float acc = js[60];
#pragma unroll
  for (int la = 0; la < KB; la++) acc += be[la] * js[la];
  joints[id] = acc;
}

// ---------------------------------------------------------------------------
// K4: kinematic chain, one wave per batch. Outputs A_rel (BS,56,12) padded
//     with a zero row (for the K=56 WMMA loop) and posed joints -> d_out.
// ---------------------------------------------------------------------------
__global__ __launch_bounds__(32) void k_chain(const float* __restrict__ rotm,
                                              const float* __restrict__ joints,
                                              float* __restrict__ arel,
                                              float* __restrict__ pj_out)
{
  int b = blockIdx.x;
  int t = threadIdx.x;
  __shared__ float R[NJ * 9], jnt[NJ * 3], rel[NJ * 3], A[NJ * 12];
  for (int e = t; e < NJ * 9; e += 32) R[e] = rotm[(size_t)b * NJ * 9 + e];
  for (int e = t; e < NJ * 3; e += 32) jnt[e] = joints[(size_t)b * NJ * 3 + e];
  __syncthreads();
  for (int e = t; e < NJ * 3; e += 32) {
    int j = e / 3, cc = e - j * 3;
    rel[e] = (j == 0) ? jnt[e] : (jnt[e] - jnt[c_par[j] * 3 + cc]);
  }
  __syncthreads();
  if (t < 12) {
    int m = t >> 2, n = t & 3;
    A[t] = (n < 3) ? R[m * 3 + n] : rel[m];
  }
  __syncthreads();
  for (int j = 1; j < NJ; j++) {
    int par = c_par[j];
    float s = 0.f;
    if (t < 12) {
      int m = t >> 2, n = t & 3;
      s = (n == 3) ? A[par * 12 + m * 4 + 3] : 0.f;
#pragma unroll
      for (int k = 0; k < 3; k++) {
        float tl = (n < 3) ? R[j * 9 + k * 3 + n] : rel[j * 3 + k];
        s += A[par * 12 + m * 4 + k] * tl;
      }
    }
    __syncthreads();
    if (t < 12) A[j * 12 + t] = s;
    __syncthreads();
  }
  for (int e = t; e < NJ * 3; e += 32) {            // posed joints (output 2)
    int j = e / 3, cc = e - j * 3;
    pj_out[(size_t)b * NJ * 3 + e] = A[j * 12 + cc * 4 + 3];
  }
  float* ab = arel + (size_t)b * 56 * 12;           // A_rel = [Rj | t - Rj*jnt]
  for (int e = t; e < NJ * 12; e += 32) {
    int j = e / 12, mn = e - j * 12;
    int m = mn >> 2, n = mn & 3;
    float val = A[e];
    if (n == 3) {
      val -= A[j*12 + m*4 + 0] * jnt[j*3 + 0]
           + A[j*12 + m*4 + 1] * jnt[j*3 + 1]
           + A[j*12 + m*4 + 2] * jnt[j*3 + 2];
    }
    ab[e] = val;
  }
  if (t < 12) ab[NJ * 12 + t] = 0.f;                // zero pad row 55
}

// ---------------------------------------------------------------------------
// K5: fused blendshape GEMM + LBS GEMM + affine apply.
//     4 waves per workgroup; each wave owns a 16-vertex subtile of a
//     64-vertex block; all 4 share one LDS-staged (16-batch) pf/betas tile.
//     All matrix math via V_WMMA_F32_16X16X4_F32.
// ---------------------------------------------------------------------------
__global__ __launch_bounds__(128) void k_verts(
    const float* __restrict__ betas, const float* __restrict__ pf,
    const float* __restrict__ sdT,   const float* __restrict__ pdirs,
    const float* __restrict__ vtmpl, const float* __restrict__ lbsw,
    const float* __restrict__ arel,  float* __restrict__ verts)
{
  const int tid  = threadIdx.x;
  const int lane = tid & 31;
  const int wv   = tid >> 5;          // wave id 0..3
  const int L16  = lane & 15;
  const int Lh   = (lane >> 4) & 1;
  const int v0   = blockIdx.x * 64 + wv * 16;   // vertex subtile base
  const int b0   = blockIdx.y * 16;             // batch tile base
  const int n0   = v0 * 3;                      // 48 vertex-component columns

  __shared__ float pfs[16 * PFS];     // staged pose_feature tile (padded rows)
  __shared__ float bet[16 * KB];      // staged betas tile
  __shared__ float vpbuf[4][16 * 48]; // per-wave v_posed tile [batch][n]
  __shared__ float Tbuf[4][16 * 16];  // per-wave T tile [v_local][col]

  // ---- stage pf + betas tiles (batch-tile rows are contiguous in ws) ----
  {
    const float* src = pf + (size_t)b0 * KPF;
    for (int i = tid; i < 16 * KPF; i += 128)
      pfs[(i / KPF) * PFS + (i % KPF)] = src[i];
    const float* bsrc = betas + (size_t)b0 * KB;
    for (int i = tid; i < 16 * KB; i += 128)
      bet[i] = bsrc[i];
  }
  __syncthreads();

  const int ncl0 = imin(n0 +  0 + L16, N3 - 1);
  const int ncl1 = imin(n0 + 16 + L16, N3 - 1);
  const int ncl2 = imin(n0 + 32 + L16, N3 - 1);

  v8f c0 = {0.f,0.f,0.f,0.f,0.f,0.f,0.f,0.f};
  v8f c1 = {0.f,0.f,0.f,0.f,0.f,0.f,0.f,0.f};
  v8f c2 = {0.f,0.f,0.f,0.f,0.f,0.f,0.f,0.f};

  // ---- segment 1: betas (K=60 exact) x active shapedirs ----
  {
    const float* bp = &bet[L16 * KB];
    const float* r0 = sdT + (size_t)(2 * Lh) * N3;
    for (int kk = 0; kk < KB; kk += 4) {
      v2f a; a.x = bp[kk + 2*Lh]; a.y = bp[kk + 2*Lh + 1];
      v2f bv;
      bv.x = r0[ncl0]; bv.y = r0[N3 + ncl0]; c0 = WMMA4(a, bv, c0);
      bv.x = r0[ncl1]; bv.y = r0[N3 + ncl1]; c1 = WMMA4(a, bv, c1);
      bv.x = r0[ncl2]; bv.y = r0[N3 + ncl2]; c2 = WMMA4(a, bv, c2);
      r0 += 4 * (size_t)N3;
    }
  }
  // ---- segment 2: pose_feature x posedirs; clamp-free main loop + peel ----
  {
    const float* pp = &pfs[L16 * PFS];
    const float* r0 = pdirs + (size_t)(2 * Lh) * N3;
    for (int kk = 0; kk < 484; kk += 4) {       // rows k,k+1 <= 483 < 486
      int k = kk + 2 * Lh;
      v2f a; a.x = pp[k]; a.y = pp[k + 1];
      v2f bv;
      bv.x = r0[ncl0]; bv.y = r0[N3 + ncl0]; c0 = WMMA4(a, bv, c0);
      bv.x = r0[ncl1]; bv.y = r0[N3 + ncl1]; c1 = WMMA4(a, bv, c1);
      bv.x = r0[ncl2]; bv.y = r0[N3 + ncl2]; c2 = WMMA4(a, bv, c2);
      r0 += 4 * (size_t)N3;
    }
    { // peeled kk = 484: rows 486/487 don't exist; A pad cols are zero
      int k = 484 + 2 * Lh;
      v2f a; a.x = pp[k]; a.y = pp[k + 1];
      const float* q0 = pdirs + (size_t)imin(k,     485) * N3;
      const float* q1 = pdirs + (size_t)imin(k + 1, 485) * N3;
      v2f bv;
      bv.x = q0[ncl0]; bv.y = q1[ncl0]; c0 = WMMA4(a, bv, c0);
      bv.x = q0[ncl1]; bv.y = q1[ncl1]; c1 = WMMA4(a, bv, c1);
      bv.x = q0[ncl2]; bv.y = q1[ncl2]; c2 = WMMA4(a, bv, c2);
    }
  }
  // ---- stage v_posed = v_template + blendshapes in LDS ----
  {
    float t0 = vtmpl[ncl0], t1 = vtmpl[ncl1], t2 = vtmpl[ncl2];
#pragma unroll
    for (int r = 0; r < 8; r++) {
      int m = r + 8 * Lh;                        // batch_local
      vpbuf[wv][m * 48 +  0 + L16] = c0[r] + t0;
      vpbuf[wv][m * 48 + 16 + L16] = c1[r] + t1;
      vpbuf[wv][m * 48 + 32 + L16] = c2[r] + t2;
    }
  }
  __syncthreads();

  // ---- LBS: per batch, T(16v x 12) = w(16v x 55) @ A_rel(55 x 12), K=56 ----
  const int vv = imin(v0 + L16, NVRT - 1);
  const float* wrow = lbsw + (size_t)vv * NJ;
  const int colc = imin(L16, 11);

  v2f aw[14];                                    // hoist A operands (constant
#pragma unroll                                   // across the 16-batch loop)
  for (int s = 0; s < 14; s++) {
    int k = s * 4 + 2 * Lh;
    aw[s].x = (k     < NJ) ? wrow[k]     : 0.f;
    aw[s].y = (k + 1 < NJ) ? wrow[k + 1] : 0.f;
  }

  for (int bb = 0; bb < 16; ++bb) {
    int b = b0 + bb;
    const float* ab = arel + (size_t)b * 56 * 12;
    v8f ct = {0.f,0.f,0.f,0.f,0.f,0.f,0.f,0.f};
#pragma unroll
    for (int s = 0; s < 14; s++) {
      int k = s * 4 + 2 * Lh;
      v2f bv;
      bv.x = ab[k * 12 + colc];                  // row 55 is zero-padded
      bv.y = ab[(k + 1) * 12 + colc];
      ct = WMMA4(aw[s], bv, ct);
    }
#pragma unroll
    for (int r = 0; r < 8; r++) Tbuf[wv][(r + 8 * Lh) * 16 + L16] = ct[r];
    __syncthreads();
    for (int e = lane; e < 48; e += 32) {        // 16 verts x 3 comps
      int vl = e / 3, cc = e - vl * 3;
      const float* tr = &Tbuf[wv][vl * 16 + cc * 4];
      const float* vp = &vpbuf[wv][bb * 48 + vl * 3];
      float o = tr[0] * vp[0] + tr[1] * vp[1] + tr[2] * vp[2] + tr[3];
      int vg = v0 + vl;
      if (vg < NVRT) verts[((size_t)b * NVRT + vg) * 3 + cc] = o;
    }
    __syncthreads();
  }
}

// ---------------------------------------------------------------------------
extern "C" void kernel_launch(void* const* d_in, const int* in_sizes, int n_in,
                              void* d_out, int out_size, void* d_ws, size_t ws_size,
                              hipStream_t stream)
{
  (void)in_sizes; (void)n_in; (void)out_size; (void)ws_size;
  const float* pose = (const float*)d_in[0];
  const float* shp  = (const float*)d_in[1];
  const float* expr = (const float*)d_in[2];
  const float* go   = (const float*)d_in[3];
  const float* vt   = (const float*)d_in[4];
  const float* sd   = (const float*)d_in[5];
  const float* pd   = (const float*)d_in[6];
  const float* jreg = (const float*)d_in[7];
  const float* lbsw = (const float*)d_in[8];

  float* ws     = (float*)d_ws;
  float* rotm   = ws;             // 256*55*9   = 126720
  float* pf     = ws + 126720;    // 256*488    = 124928
  float* betas  = ws + 251648;    // 256*60     = 15360
  float* JS     = ws + 267008;    // 165*61     = 10065 (padded 10240)
  float* joints = ws + 277248;    // 256*165    = 42240
  float* arel   = ws + 319488;    // 256*56*12  = 172032
  float* sdT    = ws + 491520;    // 60*31425   = 1885500  (total ~9.5 MB)

  float* verts = (float*)d_out;
  float* pjout = verts + (size_t)BS * NVRT * 3;

  k_rodrigues<<<(BS * NJ + 255) / 256, 256, 0, stream>>>(pose, go, rotm, pf);
  k_betas    <<<(BS * KB + 255) / 256, 256, 0, stream>>>(shp, expr, betas);
  k_sdT      <<<(KB * N3 + 255) / 256, 256, 0, stream>>>(sd, sdT);
  k_JS       <<<165, 128, 0, stream>>>(jreg, sdT, vt, JS);
  k_joints   <<<(BS * 165 + 255) / 256, 256, 0, stream>>>(betas, JS, joints);
  k_chain    <<<BS, 32, 0, stream>>>(rotm, joints, arel, pjout);
  dim3 g5((NVRT + 63) / 64, BS / 16);
  k_verts    <<<g5, 128, 0, stream>>>(betas, pf, sdT, pd, vt, lbsw, arel, verts);
}